// iSQRTCOVPool_6983616823742
// MI455X (gfx1250) — compile-verified
//
#include <hip/hip_runtime.h>
#include <hip/hip_bf16.h>
#include <math.h>

// ---------------------------------------------------------------------------
// iSQRT-COV pooling for MI455X (gfx1250, wave32, WMMA f32 16x16x4).
//   sigma = (1/n) * centered covariance  (B=64, C=128, n=3136)
//   5 Newton-Schulz iterations -> matrix sqrt, triu extraction.
// fp32 WMMA everywhere (NS is numerically sensitive; reference is fp32).
// Symmetry exploited in BOTH phases: only upper-triangular 16x16 tiles are
// computed (36 of 64), mirrored on store -> ~1.7x fewer WMMA issues.
// Tile split: 4 unconditional tiles/wave (branch-free hot loop) + a 5th tile
// for waves 0-3 in its own k-sweep (one guard per sweep, dual accumulators).
// ---------------------------------------------------------------------------

#define B_DIM     64
#define C_DIM     128
#define N_SPATIAL 3136
#define TRIU_LEN  (C_DIM * (C_DIM + 1) / 2)   // 8256
#define NTILES    36                           // upper-tri 16x16 tiles of 128x128

typedef __attribute__((ext_vector_type(2))) float v2f;
typedef __attribute__((ext_vector_type(8))) float v8f;

// decode linear upper-triangular tile index q (row-major, j>=i, 8x8 grid)
__device__ __forceinline__ void decode_triu(int q, int& io, int& jo) {
    // offset(i) = i*(17-i)/2 ; invert with fp, then fix up rounding
    int i = (int)floorf((17.0f - sqrtf(289.0f - 8.0f * (float)q)) * 0.5f);
    int off = (i * (17 - i)) / 2;
    if (off > q) { --i; off = (i * (17 - i)) / 2; }
    else {
        const int offn = ((i + 1) * (16 - i)) / 2;
        if (offn <= q) { ++i; off = offn; }
    }
    io = i;
    jo = i + (q - off);
}

// ======================= covariance (Gram) kernel ==========================
#define SPLIT            8
#define CHUNK            56                   // one image row (W)
#define CHUNKS_PER_SLICE 7                    // 8*7*56 = 3136
#define CSTR             60                   // padded LDS row stride (4r mod 64 banks)

__global__ __launch_bounds__(256) void cov_kernel(const float* __restrict__ x,
                                                  float* __restrict__ Sg,
                                                  float* __restrict__ sums) {
    __shared__ float lds[C_DIM * CSTR];

    const int b     = blockIdx.x / SPLIT;
    const int slice = blockIdx.x % SPLIT;
    const int tid   = threadIdx.x;
    const int wave  = tid >> 5;
    const int lane  = tid & 31;
    const int rsel  = lane & 15;              // row (A) / col (B) within tile
    const int koff  = (lane >> 4) * 2;        // lanes 16..31 hold K=2,3
    const int mhalf = 8 * (lane >> 4);
    const bool has5 = (wave < 4);             // waves 0-3 own a 5th tile

    // per-wave upper-tri tile list: q = wave + 8t (t=0..3 all waves, t=4 if has5)
    int ti_[5], tj_[5], aoff[5], boff[5];
#pragma unroll
    for (int t = 0; t < 5; ++t) {
        int q = wave + t * 8;
        if (q > NTILES - 1) q = NTILES - 1;   // keep decode well-defined
        int i, j;
        decode_triu(q, i, j);
        ti_[t] = i;
        tj_[t] = j;
        aoff[t] = (i * 16 + rsel) * CSTR;
        boff[t] = (j * 16 + rsel) * CSTR;
    }

    const v8f vzero = {};
    v8f acc[4], acc4a = vzero, acc4b = vzero;
#pragma unroll
    for (int t = 0; t < 4; ++t) acc[t] = vzero;
    float rowsum = 0.0f;

    const float* xb = x + (size_t)b * C_DIM * N_SPATIAL;

    for (int ch = 0; ch < CHUNKS_PER_SLICE; ++ch) {
        const int n0 = (slice * CHUNKS_PER_SLICE + ch) * CHUNK;

        // stage 128 channels x 56 floats into LDS (coalesced float4 loads)
        for (int idx = tid; idx < C_DIM * (CHUNK / 4); idx += 256) {
            const int c = idx / (CHUNK / 4);
            const int q = idx % (CHUNK / 4);
            const float4 v = *(const float4*)(xb + (size_t)c * N_SPATIAL + n0 + q * 4);
            float* dst = &lds[c * CSTR + q * 4];
            dst[0] = v.x; dst[1] = v.y; dst[2] = v.z; dst[3] = v.w;
        }

        // prefetch next chunk into cache while we compute on this one
        if (ch + 1 < CHUNKS_PER_SLICE) {
            const int n1 = n0 + CHUNK;
            for (int idx = tid; idx < C_DIM * 4; idx += 256) {
                const int c   = idx >> 2;
                const int seg = idx & 3;
                __builtin_prefetch(xb + (size_t)c * N_SPATIAL + n1 + seg * 16, 0, 3);
            }
        }
        __syncthreads();

        // fused per-channel sums (vectorized b128 reads of one LDS row)
        if (tid < C_DIM) {
            const float4* row = (const float4*)&lds[tid * CSTR];
            float s = 0.0f;
#pragma unroll
            for (int q = 0; q < CHUNK / 4; ++q) {
                const float4 v = row[q];
                s += (v.x + v.y) + (v.z + v.w);
            }
            rowsum += s;
        }

        // branch-free hot loop: 4 tiles per wave, batched fragment loads
        for (int ks = 0; ks < CHUNK / 4; ++ks) {
            const int kb = ks * 4 + koff;
#pragma unroll
            for (int t = 0; t < 4; ++t) {
                v2f a, bf;
                a.x  = lds[aoff[t] + kb];
                a.y  = lds[aoff[t] + kb + 1];
                bf.x = lds[boff[t] + kb];
                bf.y = lds[boff[t] + kb + 1];
                acc[t] = __builtin_amdgcn_wmma_f32_16x16x4_f32(
                    false, a, false, bf, (short)0, acc[t], false, false);
            }
        }
        // 5th tile: one guard per chunk, dual accumulators for 2 indep chains
        if (has5) {
            for (int ks = 0; ks < CHUNK / 4; ks += 2) {
                const int kb0 = ks * 4 + koff;
                const int kb1 = kb0 + 4;
                v2f a0, b0, a1, b1;
                a0.x = lds[aoff[4] + kb0];  a0.y = lds[aoff[4] + kb0 + 1];
                b0.x = lds[boff[4] + kb0];  b0.y = lds[boff[4] + kb0 + 1];
                a1.x = lds[aoff[4] + kb1];  a1.y = lds[aoff[4] + kb1 + 1];
                b1.x = lds[boff[4] + kb1];  b1.y = lds[boff[4] + kb1 + 1];
                acc4a = __builtin_amdgcn_wmma_f32_16x16x4_f32(
                    false, a0, false, b0, (short)0, acc4a, false, false);
                acc4b = __builtin_amdgcn_wmma_f32_16x16x4_f32(
                    false, a1, false, b1, (short)0, acc4b, false, false);
            }
        }
        __syncthreads();
    }

    // merge partial Gram: D-layout VGPR v -> m = 16*i + v + 8*(lane>=16)
    float* Sb = Sg + (size_t)b * C_DIM * C_DIM;
#pragma unroll
    for (int t = 0; t < 4; ++t) {
        const int i = ti_[t], j = tj_[t];
#pragma unroll
        for (int v = 0; v < 8; ++v) {
            const int m = i * 16 + v + mhalf;
            const int n = j * 16 + rsel;
            const float val = acc[t][v];
            atomicAdd(&Sb[m * C_DIM + n], val);
            if (i != j) atomicAdd(&Sb[n * C_DIM + m], val);      // symmetry
        }
    }
    if (has5) {
        const int i = ti_[4], j = tj_[4];
#pragma unroll
        for (int v = 0; v < 8; ++v) {
            const int m = i * 16 + v + mhalf;
            const int n = j * 16 + rsel;
            const float val = acc4a[v] + acc4b[v];
            atomicAdd(&Sb[m * C_DIM + n], val);
            if (i != j) atomicAdd(&Sb[n * C_DIM + m], val);
        }
    }
    if (tid < C_DIM) atomicAdd(&sums[b * C_DIM + tid], rowsum);
}

// ========================= Newton-Schulz kernel ============================
// One block per batch; Y, Z, T live in LDS (~198 KB < 320 KB/WGP). All NS
// iterates are polynomials in A -> symmetric; compute upper tiles only and
// mirror-store (diagonal tiles not mirrored => deterministic stores).
#define STR 132   // padded stride: 4r mod 64 -> conflict-free fragment gathers

// acc[0..3]: tiles t=0..3 (all waves); acc[4]: tile 4, valid iff has5.
__device__ __forceinline__ void wave_mm_sym(const float* __restrict__ A,
                                            const float* __restrict__ Bm,
                                            const int (&arow)[5], const int (&bcol)[5],
                                            bool has5, int koff, v8f (&acc)[5]) {
    const v8f vzero = {};
#pragma unroll
    for (int t = 0; t < 5; ++t) acc[t] = vzero;
    for (int ks = 0; ks < 32; ++ks) {
        const int kb = ks * 4 + koff;
#pragma unroll
        for (int t = 0; t < 4; ++t) {
            v2f a, bf;
            a.x  = A[arow[t] + kb];
            a.y  = A[arow[t] + kb + 1];
            bf.x = Bm[kb * STR + bcol[t]];
            bf.y = Bm[(kb + 1) * STR + bcol[t]];
            acc[t] = __builtin_amdgcn_wmma_f32_16x16x4_f32(
                false, a, false, bf, (short)0, acc[t], false, false);
        }
    }
    if (has5) {
        const v8f vz = {};
        v8f p0 = vz, p1 = vz;
        for (int ks = 0; ks < 32; ks += 2) {
            const int kb0 = ks * 4 + koff;
            const int kb1 = kb0 + 4;
            v2f a0, b0, a1, b1;
            a0.x = A[arow[4] + kb0];        a0.y = A[arow[4] + kb0 + 1];
            b0.x = Bm[kb0 * STR + bcol[4]]; b0.y = Bm[(kb0 + 1) * STR + bcol[4]];
            a1.x = A[arow[4] + kb1];        a1.y = A[arow[4] + kb1 + 1];
            b1.x = Bm[kb1 * STR + bcol[4]]; b1.y = Bm[(kb1 + 1) * STR + bcol[4]];
            p0 = __builtin_amdgcn_wmma_f32_16x16x4_f32(
                false, a0, false, b0, (short)0, p0, false, false);
            p1 = __builtin_amdgcn_wmma_f32_16x16x4_f32(
                false, a1, false, b1, (short)0, p1, false, false);
        }
#pragma unroll
        for (int v = 0; v < 8; ++v) acc[4][v] = p0[v] + p1[v];
    }
}

__device__ __forceinline__ void store_sym_tiles(float* __restrict__ O,
                                                const int (&ti_)[5], const int (&tj_)[5],
                                                bool has5, int rsel, int mhalf,
                                                const v8f (&acc)[5]) {
#pragma unroll
    for (int t = 0; t < 4; ++t) {
        const int i = ti_[t], j = tj_[t];
#pragma unroll
        for (int v = 0; v < 8; ++v) {
            const int m = i * 16 + v + mhalf;
            const int n = j * 16 + rsel;
            const float val = acc[t][v];
            O[m * STR + n] = val;
            if (i != j) O[n * STR + m] = val;   // mirror (symmetric result)
        }
    }
    if (has5) {
        const int i = ti_[4], j = tj_[4];
#pragma unroll
        for (int v = 0; v < 8; ++v) {
            const int m = i * 16 + v + mhalf;
            const int n = j * 16 + rsel;
            const float val = acc[4][v];
            O[m * STR + n] = val;
            if (i != j) O[n * STR + m] = val;
        }
    }
}

__global__ __launch_bounds__(256) void ns_kernel(const float* __restrict__ Sg,
                                                 const float* __restrict__ sums,
                                                 float* __restrict__ out) {
    __shared__ float Ybuf[C_DIM * STR];
    __shared__ float Zbuf[C_DIM * STR];
    __shared__ float Tbuf[C_DIM * STR];
    __shared__ float svec[C_DIM];
    __shared__ float red[C_DIM];
    __shared__ float trsh;

    const int b     = blockIdx.x;
    const int tid   = threadIdx.x;
    const int wave  = tid >> 5;
    const int lane  = tid & 31;
    const int rsel  = lane & 15;
    const int koff  = (lane >> 4) * 2;
    const int mhalf = 8 * (lane >> 4);
    const bool has5 = (wave < 4);
    const float inv_n = 1.0f / (float)N_SPATIAL;

    int ti_[5], tj_[5], arow[5], bcol[5];
#pragma unroll
    for (int t = 0; t < 5; ++t) {
        int q = wave + t * 8;
        if (q > NTILES - 1) q = NTILES - 1;
        int i, j;
        decode_triu(q, i, j);
        ti_[t] = i;
        tj_[t] = j;
        arow[t] = (i * 16 + rsel) * STR;
        bcol[t] = j * 16 + rsel;
    }

    if (tid < C_DIM) svec[tid] = sums[b * C_DIM + tid];
    __syncthreads();

    // sigma = (S - s s^T / n) / n ; Y = sigma (scaled below), Z = I
    const float* Sb = Sg + (size_t)b * C_DIM * C_DIM;
    for (int idx = tid; idx < C_DIM * C_DIM; idx += 256) {
        const int r = idx >> 7;
        const int c = idx & (C_DIM - 1);
        const float sig = (Sb[idx] - svec[r] * svec[c] * inv_n) * inv_n;
        Ybuf[r * STR + c] = sig;
        Zbuf[r * STR + c] = (r == c) ? 1.0f : 0.0f;
    }
    __syncthreads();

    // trace(sigma)
    if (tid < C_DIM) red[tid] = Ybuf[tid * STR + tid];
    __syncthreads();
    if (tid == 0) {
        float t = 0.0f;
        for (int i = 0; i < C_DIM; ++i) t += red[i];
        trsh = t;
    }
    __syncthreads();
    const float tr = trsh;
    const float invtr = 1.0f / tr;
    for (int idx = tid; idx < C_DIM * C_DIM; idx += 256) {
        Ybuf[(idx >> 7) * STR + (idx & (C_DIM - 1))] *= invtr;
    }
    __syncthreads();

    v8f acc[5];
    for (int it = 0; it < 5; ++it) {
        // T = 0.5*(3I - Z@Y)   (upper tiles + mirror)
        wave_mm_sym(Zbuf, Ybuf, arow, bcol, has5, koff, acc);
#pragma unroll
        for (int t = 0; t < 5; ++t) {
            const bool live = (t < 4) || has5;
            if (live) {
                const int i = ti_[t], j = tj_[t];
#pragma unroll
                for (int v = 0; v < 8; ++v) {
                    const int m = i * 16 + v + mhalf;
                    const int n = j * 16 + rsel;
                    float val = -0.5f * acc[t][v];
                    if (m == n) val += 1.5f;
                    Tbuf[m * STR + n] = val;
                    if (i != j) Tbuf[n * STR + m] = val;
                }
            }
        }
        __syncthreads();

        // Y' = Y @ T
        wave_mm_sym(Ybuf, Tbuf, arow, bcol, has5, koff, acc);
        __syncthreads();                       // all waves done reading Y
        store_sym_tiles(Ybuf, ti_, tj_, has5, rsel, mhalf, acc);
        __syncthreads();

        // Z' = T @ Z  (dead on the final iteration -> skipped)
        if (it < 4) {
            wave_mm_sym(Tbuf, Zbuf, arow, bcol, has5, koff, acc);
            __syncthreads();                   // all waves done reading Z
            store_sym_tiles(Zbuf, ti_, tj_, has5, rsel, mhalf, acc);
            __syncthreads();
        }
    }

    // c_mat = Y * sqrt(tr); emit upper triangle row by row
    const float scale = sqrtf(tr);
    float* ob = out + (size_t)b * TRIU_LEN;
    int base = 0;
    for (int r = 0; r < C_DIM; ++r) {
        const int len = C_DIM - r;
        for (int j = tid; j < len; j += 256) {
            ob[base + j] = scale * Ybuf[r * STR + r + j];
        }
        base += len;
    }
}

// =============================== launcher ==================================
extern "C" void kernel_launch(void* const* d_in, const int* in_sizes, int n_in,
                              void* d_out, int out_size, void* d_ws, size_t ws_size,
                              hipStream_t stream) {
    const float* x = (const float*)d_in[0];
    float* Sg   = (float*)d_ws;                                  // [B,128,128]
    float* sums = (float*)d_ws + (size_t)B_DIM * C_DIM * C_DIM;  // [B,128]
    float* out  = (float*)d_out;

    const size_t zero_bytes =
        ((size_t)B_DIM * C_DIM * C_DIM + (size_t)B_DIM * C_DIM) * sizeof(float);
    hipMemsetAsync(d_ws, 0, zero_bytes, stream);

    cov_kernel<<<B_DIM * SPLIT, 256, 0, stream>>>(x, Sg, sums);
    ns_kernel<<<B_DIM, 256, 0, stream>>>(Sg, sums, out);
}